// environment_network_50749333569732
// MI455X (gfx1250) — compile-verified
//
#include <hip/hip_runtime.h>
#include <math.h>

#define N_NODES 100000
#define N_EDGES 10000
#define N_PAIRS 1600000
#define F 128
#define M_TILES ((N_NODES + 15) / 16)   // 6250
#define LDS_STRIDE 132                  // 128 + 4 pad -> conflict-free ds_load_b64

typedef float v2f __attribute__((ext_vector_type(2)));
typedef float v8f __attribute__((ext_vector_type(8)));

__device__ __forceinline__ float gelu_exact(float v) {
    return 0.5f * v * (1.0f + erff(v * 0.70710678118654752f));
}

// ---------------------------------------------------------------------------
// 0) zero the CSR count arrays (only state that needs re-init per launch)
// ---------------------------------------------------------------------------
__global__ void zero_kernel(int* __restrict__ p, int n) {
    int i = blockIdx.x * blockDim.x + threadIdx.x;
    if (i < n) p[i] = 0;
}

// ---------------------------------------------------------------------------
// 1) CSR pass A: per-edge / per-node incidence counts (int atomics)
// ---------------------------------------------------------------------------
__global__ void count_kernel(const int* __restrict__ pn, const int* __restrict__ pe,
                             int* __restrict__ node_cnt, int* __restrict__ edge_cnt) {
    int i = blockIdx.x * blockDim.x + threadIdx.x;
    if (i < N_PAIRS) {
        atomicAdd(&edge_cnt[pe[i]], 1);
        atomicAdd(&node_cnt[pn[i]], 1);
    }
}

// ---------------------------------------------------------------------------
// 2) single-workgroup exclusive scan: cnt[n] -> off[n+1], cur[n]=off[n]
// ---------------------------------------------------------------------------
__global__ __launch_bounds__(1024) void scan_kernel(const int* __restrict__ cnt,
                                                    int* __restrict__ off,
                                                    int* __restrict__ cur, int n) {
    __shared__ int partial[1024];
    const int T = blockDim.x;
    const int tid = threadIdx.x;
    const int chunk = (n + T - 1) / T;
    const int base = tid * chunk;

    int s = 0;
    for (int i = 0; i < chunk; ++i) {
        int j = base + i;
        if (j < n) s += cnt[j];
    }
    partial[tid] = s;
    __syncthreads();
    // Hillis-Steele inclusive scan over the 1024 partials
    for (int ofs = 1; ofs < T; ofs <<= 1) {
        int v = (tid >= ofs) ? partial[tid - ofs] : 0;
        __syncthreads();
        partial[tid] += v;
        __syncthreads();
    }
    int run = (tid == 0) ? 0 : partial[tid - 1];
    for (int i = 0; i < chunk; ++i) {
        int j = base + i;
        if (j < n) {
            off[j] = run;
            cur[j] = run;
            run += cnt[j];
        }
    }
    if (tid == T - 1) off[n] = run;
}

// ---------------------------------------------------------------------------
// 3) CSR pass B: bucket-fill adjacency lists (int atomics on cursors)
//    edge_nodes[slot of edge e] = node ;  node_edges[slot of node n] = edge
// ---------------------------------------------------------------------------
__global__ void fill_kernel(const int* __restrict__ pn, const int* __restrict__ pe,
                            int* __restrict__ edge_cur, int* __restrict__ node_cur,
                            int* __restrict__ edge_nodes, int* __restrict__ node_edges) {
    int i = blockIdx.x * blockDim.x + threadIdx.x;
    if (i < N_PAIRS) {
        const int n = pn[i];
        const int e = pe[i];
        edge_nodes[atomicAdd(&edge_cur[e], 1)] = n;
        node_edges[atomicAdd(&node_cur[n], 1)] = e;
    }
}

// ---------------------------------------------------------------------------
// 4) m = gelu(x @ W_msg^T) * send   via V_WMMA_F32_16X16X4_F32
//    W staged in LDS (padded rows -> conflict-free ds_load_b64 for B frags).
// ---------------------------------------------------------------------------
__global__ __launch_bounds__(128) void gemm_msg_kernel(
    const float* __restrict__ x, const float* __restrict__ W,
    const float* __restrict__ action, float* __restrict__ m)
{
    __shared__ float Ws[128 * LDS_STRIDE];
    {   // cooperative stage: 128 threads x 32 float4
        const int t = threadIdx.x;
        const float4* src = (const float4*)(W + (long)t * F);
        #pragma unroll
        for (int c = 0; c < 32; ++c) {
            float4 v = src[c];
            float* dst = &Ws[t * LDS_STRIDE + c * 4];
            dst[0] = v.x; dst[1] = v.y; dst[2] = v.z; dst[3] = v.w;
        }
    }
    __syncthreads();

    const int lane = threadIdx.x & 31;
    const int wave = threadIdx.x >> 5;
    const int mt = blockIdx.x * 4 + wave;
    if (mt >= M_TILES) return;
    const int row0 = mt * 16;
    const int half = lane >> 4;
    const int l15  = lane & 15;

    v2f a[32];
    {
        const float* arow = x + (long)(row0 + l15) * F + half * 2;
        #pragma unroll
        for (int kt = 0; kt < 32; ++kt) {
            a[kt].x = arow[4 * kt + 0];
            a[kt].y = arow[4 * kt + 1];
        }
    }

    #pragma unroll 1
    for (int nt = 0; nt < 8; ++nt) {
        const int n0 = nt * 16;
        v8f acc = {0.f, 0.f, 0.f, 0.f, 0.f, 0.f, 0.f, 0.f};
        const float* bs = &Ws[(n0 + l15) * LDS_STRIDE + half * 2];
        #pragma unroll
        for (int kt = 0; kt < 32; ++kt) {
            v2f b;
            b.x = bs[4 * kt + 0];
            b.y = bs[4 * kt + 1];
            acc = __builtin_amdgcn_wmma_f32_16x16x4_f32(
                false, a[kt], false, b, (short)0, acc, false, false);
        }
        const int col = n0 + l15;
        #pragma unroll
        for (int r = 0; r < 8; ++r) {
            const int row = row0 + r + half * 8;
            const float send = action[row * 3 + 0] + action[row * 3 + 2];
            m[(long)row * F + col] = gelu_exact(acc[r]) * send;
        }
    }
}

// ---------------------------------------------------------------------------
// 5) edge reduce (pull, no atomics): one wave per edge
//    e_feat[e] = mean over incident nodes of m[node]
// ---------------------------------------------------------------------------
__global__ __launch_bounds__(256) void edge_reduce_kernel(
    const float* __restrict__ m, const int* __restrict__ edge_nodes,
    const int* __restrict__ edge_off, float* __restrict__ e_feat)
{
    const int lane = threadIdx.x & 31;
    const int e = blockIdx.x * 8 + (threadIdx.x >> 5);
    if (e >= N_EDGES) return;
    const int beg = edge_off[e];
    const int end = edge_off[e + 1];
    float4 acc = {0.f, 0.f, 0.f, 0.f};
    for (int i = beg; i < end; ++i) {
        const int node = edge_nodes[i];
        const float4 v = *(const float4*)(m + (long)node * F + lane * 4);
        acc.x += v.x; acc.y += v.y; acc.z += v.z; acc.w += v.w;
    }
    const float inv = 1.0f / fmaxf((float)(end - beg), 1.0f);
    acc.x *= inv; acc.y *= inv; acc.z *= inv; acc.w *= inv;
    *(float4*)(e_feat + (long)e * F + lane * 4) = acc;
}

// ---------------------------------------------------------------------------
// 6) node reduce (pull, no atomics): one wave per node
//    v_feat[n] = sum over incident edges of e_feat[edge]   (raw sum;
//    division by node degree + receive scale fused into GEMM2 epilogue)
// ---------------------------------------------------------------------------
__global__ __launch_bounds__(256) void node_reduce_kernel(
    const float* __restrict__ e_feat, const int* __restrict__ node_edges,
    const int* __restrict__ node_off, float* __restrict__ v_feat)
{
    const int lane = threadIdx.x & 31;
    const int n = blockIdx.x * 8 + (threadIdx.x >> 5);
    if (n >= N_NODES) return;
    const int beg = node_off[n];
    const int end = node_off[n + 1];
    float4 acc = {0.f, 0.f, 0.f, 0.f};
    for (int i = beg; i < end; ++i) {
        const int e = node_edges[i];
        const float4 v = *(const float4*)(e_feat + (long)e * F + lane * 4);
        acc.x += v.x; acc.y += v.y; acc.z += v.z; acc.w += v.w;
    }
    *(float4*)(v_feat + (long)n * F + lane * 4) = acc;
}

// ---------------------------------------------------------------------------
// 7) h = gelu(x @ W_upd^T + b_upd + v_feat * receive / max(node_deg,1))
// ---------------------------------------------------------------------------
__global__ __launch_bounds__(128) void gemm_upd_kernel(
    const float* __restrict__ x, const float* __restrict__ W,
    const float* __restrict__ b_upd, const float* __restrict__ action,
    const float* __restrict__ v_feat, const int* __restrict__ node_off,
    float* __restrict__ out)
{
    __shared__ float Ws[128 * LDS_STRIDE];
    {
        const int t = threadIdx.x;
        const float4* src = (const float4*)(W + (long)t * F);
        #pragma unroll
        for (int c = 0; c < 32; ++c) {
            float4 v = src[c];
            float* dst = &Ws[t * LDS_STRIDE + c * 4];
            dst[0] = v.x; dst[1] = v.y; dst[2] = v.z; dst[3] = v.w;
        }
    }
    __syncthreads();

    const int lane = threadIdx.x & 31;
    const int wave = threadIdx.x >> 5;
    const int mt = blockIdx.x * 4 + wave;
    if (mt >= M_TILES) return;
    const int row0 = mt * 16;
    const int half = lane >> 4;
    const int l15  = lane & 15;

    v2f a[32];
    {
        const float* arow = x + (long)(row0 + l15) * F + half * 2;
        #pragma unroll
        for (int kt = 0; kt < 32; ++kt) {
            a[kt].x = arow[4 * kt + 0];
            a[kt].y = arow[4 * kt + 1];
        }
    }

    float scale[8];   // receive[row] / max(deg(row),1)
    #pragma unroll
    for (int r = 0; r < 8; ++r) {
        const int row = row0 + r + half * 8;
        const float receive = action[row * 3 + 0] + action[row * 3 + 1];
        const int deg = node_off[row + 1] - node_off[row];
        scale[r] = receive / fmaxf((float)deg, 1.0f);
    }

    #pragma unroll 1
    for (int nt = 0; nt < 8; ++nt) {
        const int n0 = nt * 16;
        v8f acc = {0.f, 0.f, 0.f, 0.f, 0.f, 0.f, 0.f, 0.f};
        const float* bs = &Ws[(n0 + l15) * LDS_STRIDE + half * 2];
        #pragma unroll
        for (int kt = 0; kt < 32; ++kt) {
            v2f b;
            b.x = bs[4 * kt + 0];
            b.y = bs[4 * kt + 1];
            acc = __builtin_amdgcn_wmma_f32_16x16x4_f32(
                false, a[kt], false, b, (short)0, acc, false, false);
        }
        const int col = n0 + l15;
        const float bias = b_upd[col];
        #pragma unroll
        for (int r = 0; r < 8; ++r) {
            const int row = row0 + r + half * 8;
            const float mi = v_feat[(long)row * F + col] * scale[r];
            out[(long)row * F + col] = gelu_exact(acc[r] + bias + mi);
        }
    }
}

// ---------------------------------------------------------------------------
// launcher
// ---------------------------------------------------------------------------
extern "C" void kernel_launch(void* const* d_in, const int* in_sizes, int n_in,
                              void* d_out, int out_size, void* d_ws, size_t ws_size,
                              hipStream_t stream) {
    const float* x         = (const float*)d_in[0];
    const float* action    = (const float*)d_in[1];
    const float* W_msg     = (const float*)d_in[2];
    const float* W_upd     = (const float*)d_in[3];
    const float* b_upd     = (const float*)d_in[4];
    const int*   pair_node = (const int*)d_in[5];
    const int*   pair_edge = (const int*)d_in[6];
    float* out = (float*)d_out;

    // ---- workspace layout (v_feat aliases m: m is dead after edge_reduce) ----
    float* m      = (float*)d_ws;                         // N*F floats
    float* v_feat = m;                                    // alias
    float* e_feat = m + (size_t)N_NODES * F;              // E*F floats
    int* base_i   = (int*)(e_feat + (size_t)N_EDGES * F);
    int* edge_cnt   = base_i;                             // E
    int* node_cnt   = edge_cnt   + N_EDGES;               // N
    int* edge_off   = node_cnt   + N_NODES;               // E+1
    int* node_off   = edge_off   + (N_EDGES + 1);         // N+1
    int* edge_cur   = node_off   + (N_NODES + 1);         // E
    int* node_cur   = edge_cur   + N_EDGES;               // N
    int* edge_nodes = node_cur   + N_NODES;               // N_PAIRS
    int* node_edges = edge_nodes + N_PAIRS;               // N_PAIRS

    // CSR build
    zero_kernel<<<(N_EDGES + N_NODES + 255) / 256, 256, 0, stream>>>(edge_cnt,
                                                                     N_EDGES + N_NODES);
    count_kernel<<<(N_PAIRS + 255) / 256, 256, 0, stream>>>(pair_node, pair_edge,
                                                            node_cnt, edge_cnt);
    scan_kernel<<<1, 1024, 0, stream>>>(edge_cnt, edge_off, edge_cur, N_EDGES);
    scan_kernel<<<1, 1024, 0, stream>>>(node_cnt, node_off, node_cur, N_NODES);
    fill_kernel<<<(N_PAIRS + 255) / 256, 256, 0, stream>>>(pair_node, pair_edge,
                                                           edge_cur, node_cur,
                                                           edge_nodes, node_edges);

    // message GEMM (WMMA) + gelu*send
    gemm_msg_kernel<<<(M_TILES + 3) / 4, 128, 0, stream>>>(x, W_msg, action, m);

    // v2e mean then e2v sum (pull reductions, no float atomics)
    edge_reduce_kernel<<<(N_EDGES + 7) / 8, 256, 0, stream>>>(m, edge_nodes,
                                                              edge_off, e_feat);
    node_reduce_kernel<<<(N_NODES + 7) / 8, 256, 0, stream>>>(e_feat, node_edges,
                                                              node_off, v_feat);

    // update GEMM (WMMA) + fused epilogue
    gemm_upd_kernel<<<(M_TILES + 3) / 4, 128, 0, stream>>>(x, W_upd, b_upd, action,
                                                           v_feat, node_off, out);
}